// GCN_89386859365068
// MI455X (gfx1250) — compile-verified
//
#include <hip/hip_runtime.h>

#define DD 96
#define WT_STRIDE 98   // padded transposed-W row stride (even: keeps 8B align; odd-ish bank mapping)
#define BN_EPS 1e-5f
#define SLOPE 0.05f

typedef __attribute__((ext_vector_type(2))) float v2f;
typedef __attribute__((ext_vector_type(8))) float v8f;

// ---------------- degree / normalization ----------------

__global__ __launch_bounds__(256) void k_init_deg(float* __restrict__ deg, int n) {
    int i = blockIdx.x * 256 + threadIdx.x;
    if (i < n) deg[i] = 1.0f;   // self-loop contributes 1 to every node's degree
}

__global__ __launch_bounds__(256) void k_count_deg(const int* __restrict__ dst,
                                                   float* __restrict__ deg, int nE) {
    int i = blockIdx.x * 256 + threadIdx.x;
    if (i < nE)
        __hip_atomic_fetch_add(&deg[dst[i]], 1.0f, __ATOMIC_RELAXED, __HIP_MEMORY_SCOPE_AGENT);
}

__global__ __launch_bounds__(256) void k_rsqrt(float* __restrict__ deg, int n) {
    int i = blockIdx.x * 256 + threadIdx.x;
    if (i < n) deg[i] = rsqrtf(deg[i]);   // deg >= 1 always, matches reference dinv
}

__global__ __launch_bounds__(256) void k_zero(float* __restrict__ p, int n) {
    int i = blockIdx.x * 256 + threadIdx.x;
    if (i < n) p[i] = 0.0f;
}

// ---------------- GEMM: H[N,96] = X[N,96] @ W[96,96] via V_WMMA_F32_16X16X4_F32 ----------------
// One wave computes a 16-row strip of the output (6 tiles of 16x16, K-loop of 24 steps of 4).
// W is staged TRANSPOSED in LDS so each B fragment is one contiguous 8B ds_load_b64.

__global__ __launch_bounds__(256) void k_gemm96(const float* __restrict__ X,
                                                const float* __restrict__ W,
                                                float* __restrict__ H, int nRows) {
    __shared__ float Wt[DD * WT_STRIDE];          // Wt[col * 98 + k] = W[k * 96 + col]
    for (int i = threadIdx.x; i < DD * DD; i += 256) {
        int k = i / DD, col = i - k * DD;
        Wt[col * WT_STRIDE + k] = W[i];
    }
    __syncthreads();

    int wave = (blockIdx.x * 256 + threadIdx.x) >> 5;
    int lane = threadIdx.x & 31;
    int m0 = wave * 16;
    if (m0 >= nRows) return;             // wave-uniform: EXEC stays all-ones for WMMA

    const int lhalf = lane >> 4;         // 0 for lanes 0-15, 1 for lanes 16-31
    const int l16   = lane & 15;
    const int khalf = lhalf * 2;         // K sub-offset per half-wave (A/B layout, ISA 7.12.2)

    const float* Xrow = X + (size_t)(m0 + l16) * DD;

    v8f acc[6];
#pragma unroll
    for (int t = 0; t < 6; ++t)
#pragma unroll
        for (int r = 0; r < 8; ++r) acc[t][r] = 0.0f;

    for (int k = 0; k < DD; k += 4) {
        // A fragment 16x4: lanes 0-15 -> K={k,k+1}, lanes 16-31 -> K={k+2,k+3}
        v2f a = *(const v2f*)(Xrow + k + khalf);   // 8B aligned (row stride 384B, k+khalf even)
        int kr = k + khalf;
#pragma unroll
        for (int t = 0; t < 6; ++t) {
            int col = t * 16 + l16;
            // B fragment 4x16: contiguous {W[kr][col], W[kr+1][col]} in transposed LDS
            v2f b = *(const v2f*)(&Wt[col * WT_STRIDE + kr]);   // single ds_load_b64
            acc[t] = __builtin_amdgcn_wmma_f32_16x16x4_f32(
                false, a, false, b, (short)0, acc[t], false, false);
        }
    }

    // C/D layout: VGPR r, lanes 0-15 -> M=r, lanes 16-31 -> M=r+8; N = lane&15
    int rbase = m0 + lhalf * 8;
#pragma unroll
    for (int t = 0; t < 6; ++t) {
        int col = t * 16 + l16;
#pragma unroll
        for (int r = 0; r < 8; ++r)
            H[(size_t)(rbase + r) * DD + col] = acc[t][r];
    }
}

// ---------------- edge scatter: OUT[dst] += norm * H[src], one wave per edge ----------------

__global__ __launch_bounds__(256) void k_scatter(const float* __restrict__ H,
                                                 const float* __restrict__ dinv,
                                                 const int* __restrict__ src,
                                                 const int* __restrict__ dst,
                                                 float* __restrict__ OUT, int nE) {
    int e = (blockIdx.x * 256 + threadIdx.x) >> 5;
    if (e >= nE) return;
    int lane = threadIdx.x & 31;
    int s = src[e], d = dst[e];
    float nrm = dinv[s] * dinv[d];
    const float* hs = H + (size_t)s * DD;
    float*       od = OUT + (size_t)d * DD;
#pragma unroll
    for (int j = lane; j < DD; j += 32)
        __hip_atomic_fetch_add(&od[j], hs[j] * nrm, __ATOMIC_RELAXED, __HIP_MEMORY_SCOPE_AGENT);
}

// ---------------- layer-1 epilogue: + self-loop + bias, LeakyReLU, BatchNorm (eval) ----------------

__global__ __launch_bounds__(256) void k_post1(float* __restrict__ X1,        // agg, in/out
                                               const float* __restrict__ H,  // pre-agg h = X@W1
                                               const float* __restrict__ dinv,
                                               const float* __restrict__ b1,
                                               const float* __restrict__ g,
                                               const float* __restrict__ be,
                                               const float* __restrict__ mu,
                                               const float* __restrict__ var, int n) {
    int i = blockIdx.x * 256 + threadIdx.x;
    if (i >= n) return;
    int node = i / DD;
    int j    = i - node * DD;
    float di = dinv[node];
    float v  = X1[i] + H[i] * di * di + b1[j];      // self-loop norm = dinv^2
    v = (v >= 0.0f) ? v : SLOPE * v;
    v = (v - mu[j]) * rsqrtf(var[j] + BN_EPS) * g[j] + be[j];
    X1[i] = v;
}

// ---------------- layer-2 epilogue: + self-loop + bias ----------------

__global__ __launch_bounds__(256) void k_post2(float* __restrict__ OUT,
                                               const float* __restrict__ H,
                                               const float* __restrict__ dinv,
                                               const float* __restrict__ b2, int n) {
    int i = blockIdx.x * 256 + threadIdx.x;
    if (i >= n) return;
    int node = i / DD;
    int j    = i - node * DD;
    float di = dinv[node];
    OUT[i] = OUT[i] + H[i] * di * di + b2[j];
}

// ---------------- driver ----------------

extern "C" void kernel_launch(void* const* d_in, const int* in_sizes, int n_in,
                              void* d_out, int out_size, void* d_ws, size_t ws_size,
                              hipStream_t stream) {
    const float* X    = (const float*)d_in[0];
    const int*   ei   = (const int*)d_in[1];   // [2, E] int32
    const float* W1   = (const float*)d_in[2];
    const float* b1   = (const float*)d_in[3];
    const float* W2   = (const float*)d_in[4];
    const float* b2   = (const float*)d_in[5];
    const float* gam  = (const float*)d_in[6];
    const float* bet  = (const float*)d_in[7];
    const float* mu   = (const float*)d_in[8];
    const float* var  = (const float*)d_in[9];

    const int N = in_sizes[0] / DD;
    const int E = in_sizes[1] / 2;
    const int* srcI = ei;
    const int* dstI = ei + E;

    float* dinv = (float*)d_ws;                 // N floats
    float* hA   = dinv + N;                     // N*96: h1, reused for h2
    float* hB   = hA + (size_t)N * DD;          // N*96: x1 (post layer-1)
    float* out  = (float*)d_out;

    const int nElems    = N * DD;
    const int gridN     = (N + 255) / 256;
    const int gridE     = (E + 255) / 256;
    const int gridElems = (nElems + 255) / 256;
    const int gemmGrid  = ((N + 15) / 16 + 7) / 8;   // 8 waves per 256-thread block
    const int scatGrid  = (E + 7) / 8;               // 1 wave per edge

    // symmetric normalization dinv = rsqrt(deg), deg = in-degree + 1 (self loop)
    k_init_deg <<<gridN, 256, 0, stream>>>(dinv, N);
    k_count_deg<<<gridE, 256, 0, stream>>>(dstI, dinv, E);
    k_rsqrt    <<<gridN, 256, 0, stream>>>(dinv, N);

    // layer 1: h = X@W1 ; agg = scatter(norm*h) ; x1 = BN(leaky(agg + selfloop + b1))
    k_gemm96 <<<gemmGrid,  256, 0, stream>>>(X, W1, hA, N);
    k_zero   <<<gridElems, 256, 0, stream>>>(hB, nElems);
    k_scatter<<<scatGrid,  256, 0, stream>>>(hA, dinv, srcI, dstI, hB, E);
    k_post1  <<<gridElems, 256, 0, stream>>>(hB, hA, dinv, b1, gam, bet, mu, var, nElems);

    // layer 2: h = x1@W2 ; out = scatter(norm*h) + selfloop + b2
    k_gemm96 <<<gemmGrid,  256, 0, stream>>>(hB, W2, hA, N);
    k_zero   <<<gridElems, 256, 0, stream>>>(out, nElems);
    k_scatter<<<scatGrid,  256, 0, stream>>>(hA, dinv, srcI, dstI, out, E);
    k_post2  <<<gridElems, 256, 0, stream>>>(out, hA, dinv, b2, nElems);
}